// PPOFeatureExtractor_43748536877081
// MI455X (gfx1250) — compile-verified
//
#include <hip/hip_runtime.h>
#include <hip/hip_bf16.h>

// ---------------- problem constants (match reference) ----------------
#define NN      20000
#define EE      640000
#define F_IN    128
#define HEADS   8
#define HDIM    64
#define GAT_OUT 512        // HEADS*HDIM
#define OUT_C   576        // 512 + 64
#define LT      200000     // 10*N
#define CT      8
#define NEG_SLOPE 0.2f

typedef float v2f __attribute__((ext_vector_type(2)));
typedef float v8f __attribute__((ext_vector_type(8)));

// ---------------- fp32 WMMA helper (V_WMMA_F32_16X16X4_F32) ----------------
__device__ inline v8f wmma_f32_4(v2f a, v2f b, v8f c) {
  return __builtin_amdgcn_wmma_f32_16x16x4_f32(
      /*neg_a=*/false, a, /*neg_b=*/false, b,
      /*c_mod=*/(short)0, c, /*reuse_a=*/false, /*reuse_b=*/false);
}

// ---------------- generic elementwise fill ----------------
__global__ void fill_f32(float* __restrict__ p, float v, int n) {
  int i = blockIdx.x * blockDim.x + threadIdx.x;
  if (i < n) p[i] = v;
}

// ---------------- simple WMMA GEMM: C = A(MxK) @ B(KxN), B row-major -------
// one wave (32 threads) per 16x16 tile; used for the two small GCN/GAT GEMMs
__global__ void wmma_gemm_nn(const float* __restrict__ A, int lda,
                             const float* __restrict__ B, int ldb,
                             float* __restrict__ C, int ldc,
                             const float* __restrict__ bias, int relu, int K) {
  const int lane = threadIdx.x & 31;
  const int half = lane >> 4;        // 0: lanes 0-15, 1: lanes 16-31
  const int lm   = lane & 15;
  const int row0 = blockIdx.x * 16;
  const int col0 = blockIdx.y * 16;
  v8f acc = {};
  const float* ap = A + (size_t)(row0 + lm) * lda + half * 2;
  const float* bp = B + (size_t)(half * 2) * ldb + (col0 + lm);
  for (int k = 0; k < K; k += 4) {
    v2f a, b;
    a.x = ap[0];            // A[M][k + 2*half]     -> VGPR0 per ISA 16x4 layout
    a.y = ap[1];            // A[M][k + 2*half + 1] -> VGPR1
    b.x = bp[0];            // B[k + 2*half][col]
    b.y = bp[ldb];          // B[k + 2*half + 1][col]
    acc = wmma_f32_4(a, b, acc);
    ap += 4;
    bp += (size_t)4 * ldb;
  }
  float bv = bias ? bias[col0 + lm] : 0.0f;
  #pragma unroll
  for (int g = 0; g < 8; ++g) {      // C/D: lane half 0 -> M=g, half 1 -> M=8+g
    int row = row0 + half * 8 + g;
    float v = acc[g] + bv;
    if (relu) v = fmaxf(v, 0.0f);
    C[(size_t)row * ldc + col0 + lm] = v;
  }
}

// ---------------- LDS-tiled WMMA GEMM: C = A(MxK) @ Bt(NxK)^T ----------------
// Block: 128 threads = 4 waves in a 2x2 tile arrangement -> 32x32 output tile.
// K is consumed in KC=32 chunks staged through LDS (pitch 36 floats: 16B-aligned
// for b128 stores, and 36*r mod 64 is conflict-free over 16 rows on the 64 banks).
// A chunk loads are float4 (global_load_b128); fragments come from ds_load_b64.
// Next chunk is prefetched (global_prefetch_b8) while WMMAs run on the current one.
#define BM 32
#define BN 32
#define KC 32
#define LPITCH 36

__global__ void __launch_bounds__(128)
wmma_gemm_nt_lds(const float* __restrict__ A, int lda,
                 const float* __restrict__ Bt, int ldbt,
                 float* __restrict__ C, int ldc,
                 const float* __restrict__ bias, int relu, int K) {
  __shared__ float As[BM * LPITCH];
  __shared__ float Bs[BN * LPITCH];

  const int tid  = threadIdx.x;
  const int lane = tid & 31;
  const int wave = tid >> 5;         // 0..3
  const int wm   = wave & 1;         // M sub-tile
  const int wn   = wave >> 1;        // N sub-tile
  const int half = lane >> 4;
  const int lm   = lane & 15;
  const int row0 = blockIdx.x * BM;
  const int col0 = blockIdx.y * BN;

  // cooperative-load coordinates: 256 float4 slots per matrix, 128 threads -> 2 each
  // slot s: r = s >> 3 (8 float4 per 32-float row), c4 = s & 7
  v8f acc = {};

  for (int kc = 0; kc < K; kc += KC) {
    #pragma unroll
    for (int s = tid; s < (BM * KC) / 4; s += 128) {
      int r  = s >> 3;
      int c4 = (s & 7) * 4;
      const float* gA = A  + (size_t)(row0 + r) * lda  + kc + c4;
      const float* gB = Bt + (size_t)(col0 + r) * ldbt + kc + c4;
      *(float4*)&As[r * LPITCH + c4] = *(const float4*)gA;
      *(float4*)&Bs[r * LPITCH + c4] = *(const float4*)gB;
      if (kc + KC < K) {                       // overlap next chunk's L2->WGP latency
        __builtin_prefetch(gA + KC, 0, 3);     // -> global_prefetch_b8
        __builtin_prefetch(gB + KC, 0, 3);
      }
    }
    __syncthreads();

    const float* arow = &As[(wm * 16 + lm) * LPITCH];
    const float* brow = &Bs[(wn * 16 + lm) * LPITCH];
    #pragma unroll
    for (int k = 0; k < KC; k += 4) {
      v2f a, b;
      a.x = arow[k + 2 * half];      // ds_load_b64
      a.y = arow[k + 2 * half + 1];
      b.x = brow[k + 2 * half];      // ds_load_b64
      b.y = brow[k + 2 * half + 1];
      acc = wmma_f32_4(a, b, acc);
    }
    __syncthreads();
  }

  const int col = col0 + wn * 16 + lm;
  float bv = bias ? bias[col] : 0.0f;
  #pragma unroll
  for (int g = 0; g < 8; ++g) {
    int row = row0 + wm * 16 + half * 8 + g;
    float v = acc[g] + bv;
    if (relu) v = fmaxf(v, 0.0f);
    C[(size_t)row * ldc + col] = v;
  }
}

// ---------------- GCN ----------------
__global__ void deg_count(const long long* __restrict__ dst, float* __restrict__ deg) {
  int e = blockIdx.x * blockDim.x + threadIdx.x;
  if (e >= EE) return;
  atomicAdd(&deg[(int)dst[e]], 1.0f);
}

__global__ void gcn_aggregate(const long long* __restrict__ src,
                              const long long* __restrict__ dst,
                              const float* __restrict__ h,
                              const float* __restrict__ deg,
                              float* __restrict__ g) {
  int e = blockIdx.x * blockDim.x + threadIdx.x;
  if (e >= EE + NN) return;
  int s_, d_;
  if (e < EE) { s_ = (int)src[e]; d_ = (int)dst[e]; }
  else        { s_ = d_ = e - EE; }                 // self-loop
  float norm = rsqrtf(deg[s_]) * rsqrtf(deg[d_]);   // deg >= 1 by construction
  const float* hp = h + (size_t)s_ * 16;
  float* gp = g + (size_t)d_ * 16;
  #pragma unroll
  for (int c = 0; c < 16; ++c) atomicAdd(&gp[c], hp[c] * norm);
}

__global__ void gcn_bias_relu(float* __restrict__ g, const float* __restrict__ b1) {
  int i = blockIdx.x * blockDim.x + threadIdx.x;
  if (i >= NN * 16) return;
  g[i] = fmaxf(g[i] + b1[i & 15], 0.0f);
}

// ---------------- GAT ----------------
__global__ void gat_attn_coeff(const float* __restrict__ h2,
                               const float* __restrict__ att_s,
                               const float* __restrict__ att_d,
                               float* __restrict__ a_s, float* __restrict__ a_d) {
  int idx = blockIdx.x * blockDim.x + threadIdx.x;      // n*8 + head
  if (idx >= NN * HEADS) return;
  int n = idx >> 3, hh = idx & 7;
  const float* hp = h2 + (size_t)n * GAT_OUT + hh * HDIM;
  const float* as = att_s + hh * HDIM;
  const float* ad = att_d + hh * HDIM;
  float vs = 0.0f, vd = 0.0f;
  #pragma unroll 8
  for (int d = 0; d < HDIM; ++d) { vs = fmaf(hp[d], as[d], vs); vd = fmaf(hp[d], ad[d], vd); }
  a_s[idx] = vs; a_d[idx] = vd;
}

__device__ inline void atomicMaxF32(float* addr, float val) {
  // ordered-int trick: works for mixed signs, init with -FLT_MAX
  if (val >= 0.0f) atomicMax((int*)addr, __float_as_int(val));
  else             atomicMin((unsigned int*)addr, (unsigned int)__float_as_int(val));
}

__device__ inline void edge_endpoints(int e, const long long* src, const long long* dst,
                                      int& s_, int& d_) {
  if (e < EE) { s_ = (int)src[e]; d_ = (int)dst[e]; }
  else        { s_ = d_ = e - EE; }
}

__global__ void gat_edge_max(const long long* __restrict__ src,
                             const long long* __restrict__ dst,
                             const float* __restrict__ a_s,
                             const float* __restrict__ a_d,
                             float* __restrict__ m) {
  int e = blockIdx.x * blockDim.x + threadIdx.x;
  if (e >= EE + NN) return;
  int s_, d_; edge_endpoints(e, src, dst, s_, d_);
  #pragma unroll
  for (int hh = 0; hh < HEADS; ++hh) {
    float v = a_s[s_ * HEADS + hh] + a_d[d_ * HEADS + hh];
    v = (v >= 0.0f) ? v : NEG_SLOPE * v;                // LeakyReLU(0.2)
    atomicMaxF32(&m[d_ * HEADS + hh], v);
  }
}

__global__ void gat_edge_sum(const long long* __restrict__ src,
                             const long long* __restrict__ dst,
                             const float* __restrict__ a_s,
                             const float* __restrict__ a_d,
                             const float* __restrict__ m,
                             float* __restrict__ s) {
  int e = blockIdx.x * blockDim.x + threadIdx.x;
  if (e >= EE + NN) return;
  int s_, d_; edge_endpoints(e, src, dst, s_, d_);
  #pragma unroll
  for (int hh = 0; hh < HEADS; ++hh) {
    float v = a_s[s_ * HEADS + hh] + a_d[d_ * HEADS + hh];
    v = (v >= 0.0f) ? v : NEG_SLOPE * v;
    atomicAdd(&s[d_ * HEADS + hh], __expf(v - m[d_ * HEADS + hh]));
  }
}

__global__ void gat_scatter(const long long* __restrict__ src,
                            const long long* __restrict__ dst,
                            const float* __restrict__ a_s,
                            const float* __restrict__ a_d,
                            const float* __restrict__ m,
                            const float* __restrict__ s,
                            const float* __restrict__ h2,
                            float* __restrict__ out) {
  int idx = blockIdx.x * blockDim.x + threadIdx.x;      // edge*8 + head
  if (idx >= (EE + NN) * HEADS) return;
  int e = idx >> 3, hh = idx & 7;
  int s_, d_; edge_endpoints(e, src, dst, s_, d_);
  float v = a_s[s_ * HEADS + hh] + a_d[d_ * HEADS + hh];
  v = (v >= 0.0f) ? v : NEG_SLOPE * v;
  float ex = __expf(v - m[d_ * HEADS + hh]);
  float alpha = ex / (s[d_ * HEADS + hh] + 1e-16f);
  const float* hp = h2 + (size_t)s_ * GAT_OUT + hh * HDIM;
  float* op = out + (size_t)d_ * OUT_C + hh * HDIM;     // out[:, 0:512]
  #pragma unroll 8
  for (int d = 0; d < HDIM; ++d) atomicAdd(&op[d], hp[d] * alpha);
}

__global__ void out_bias_gat(float* __restrict__ out, const float* __restrict__ b2) {
  int idx = blockIdx.x * blockDim.x + threadIdx.x;      // n*512 + c
  if (idx >= NN * GAT_OUT) return;
  int n = idx >> 9, c = idx & 511;
  out[(size_t)n * OUT_C + c] += b2[c];
}

// ---------------- temporal conv1d (k=3, SAME, +bias, ReLU) ----------------
__global__ void conv1d_relu(const float* __restrict__ in, const float* __restrict__ w,
                            const float* __restrict__ b, float* __restrict__ out,
                            int Cin) {
  int l = blockIdx.x * blockDim.x + threadIdx.x;
  int o = blockIdx.y;
  if (l >= LT) return;
  float acc = b[o];
  for (int c = 0; c < Cin; ++c) {
    const float* ip = in + (size_t)c * LT;
    const float* wp = w + (size_t)(o * Cin + c) * 3;
    float x0 = (l > 0)      ? ip[l - 1] : 0.0f;
    float x1 = ip[l];
    float x2 = (l < LT - 1) ? ip[l + 1] : 0.0f;
    acc = fmaf(wp[0], x0, acc);
    acc = fmaf(wp[1], x1, acc);
    acc = fmaf(wp[2], x2, acc);
  }
  out[(size_t)o * LT + l] = fmaxf(acc, 0.0f);
}

// ---------------- launcher ----------------
extern "C" void kernel_launch(void* const* d_in, const int* in_sizes, int n_in,
                              void* d_out, int out_size, void* d_ws, size_t ws_size,
                              hipStream_t stream) {
  const float*     x       = (const float*)d_in[0];
  const long long* eidx    = (const long long*)d_in[1];   // int64 [2, E]
  const float*     temporal= (const float*)d_in[2];
  const float*     W1      = (const float*)d_in[3];
  const float*     b1      = (const float*)d_in[4];
  const float*     W2      = (const float*)d_in[5];
  const float*     att_src = (const float*)d_in[6];
  const float*     att_dst = (const float*)d_in[7];
  const float*     b2      = (const float*)d_in[8];
  const float*     cw1     = (const float*)d_in[9];
  const float*     cb1     = (const float*)d_in[10];
  const float*     cw2     = (const float*)d_in[11];
  const float*     cb2     = (const float*)d_in[12];
  const float*     cw3     = (const float*)d_in[13];
  const float*     cb3     = (const float*)d_in[14];
  const float*     fw1     = (const float*)d_in[15];      // [256, 1280]
  const float*     fb1     = (const float*)d_in[16];
  const float*     fw2     = (const float*)d_in[17];      // [64, 256]
  const float*     fb2     = (const float*)d_in[18];

  const long long* srcp = eidx;
  const long long* dstp = eidx + EE;
  float* out = (float*)d_out;

  // ---- workspace carve-out (floats) ----
  float* ws = (float*)d_ws;
  float* h    = ws;                 ws += (size_t)NN * 16;        // GCN gemm out
  float* g    = ws;                 ws += (size_t)NN * 16;        // GCN activated
  float* deg  = ws;                 ws += NN;
  float* h2   = ws;                 ws += (size_t)NN * GAT_OUT;   // GAT lin out
  float* a_s  = ws;                 ws += NN * HEADS;
  float* a_d  = ws;                 ws += NN * HEADS;
  float* mbuf = ws;                 ws += NN * HEADS;
  float* sbuf = ws;                 ws += NN * HEADS;
  float* t1   = ws;                 ws += (size_t)32  * LT;       // conv1 out (reused for u)
  float* t2   = ws;                 ws += (size_t)64  * LT;       // conv2 out
  float* t3   = ws;                 ws += (size_t)128 * LT;       // conv3 out == t[20000,1280]
  float* u    = t1;                 // FC1 out [20000,256]; t1 is dead after conv2

  const int TB = 256;
  dim3 blk(TB), wblk(32), tblk(128);

  // ---- init ----
  fill_f32<<<(NN * 16 + TB - 1) / TB, blk, 0, stream>>>(g, 0.0f, NN * 16);
  fill_f32<<<(NN + TB - 1) / TB, blk, 0, stream>>>(deg, 1.0f, NN);            // self-loop
  fill_f32<<<(NN * HEADS + TB - 1) / TB, blk, 0, stream>>>(mbuf, -3.402823466e38f, NN * HEADS);
  fill_f32<<<(NN * HEADS + TB - 1) / TB, blk, 0, stream>>>(sbuf, 0.0f, NN * HEADS);
  fill_f32<<<(NN * OUT_C + TB - 1) / TB, blk, 0, stream>>>(out, 0.0f, NN * OUT_C);

  // ---- graph branch ----
  // h = x @ W1   (20000x128x16, fp32 WMMA)
  wmma_gemm_nn<<<dim3(NN / 16, 16 / 16), wblk, 0, stream>>>(
      x, F_IN, W1, 16, h, 16, nullptr, 0, F_IN);
  deg_count<<<(EE + TB - 1) / TB, blk, 0, stream>>>(dstp, deg);
  gcn_aggregate<<<(EE + NN + TB - 1) / TB, blk, 0, stream>>>(srcp, dstp, h, deg, g);
  gcn_bias_relu<<<(NN * 16 + TB - 1) / TB, blk, 0, stream>>>(g, b1);

  // h2 = g @ W2  (20000x16x512, fp32 WMMA)
  wmma_gemm_nn<<<dim3(NN / 16, GAT_OUT / 16), wblk, 0, stream>>>(
      g, 16, W2, GAT_OUT, h2, GAT_OUT, nullptr, 0, 16);
  gat_attn_coeff<<<(NN * HEADS + TB - 1) / TB, blk, 0, stream>>>(h2, att_src, att_dst, a_s, a_d);
  gat_edge_max<<<(EE + NN + TB - 1) / TB, blk, 0, stream>>>(srcp, dstp, a_s, a_d, mbuf);
  gat_edge_sum<<<(EE + NN + TB - 1) / TB, blk, 0, stream>>>(srcp, dstp, a_s, a_d, mbuf, sbuf);
  gat_scatter<<<((EE + NN) * HEADS + TB - 1) / TB, blk, 0, stream>>>(
      srcp, dstp, a_s, a_d, mbuf, sbuf, h2, out);
  out_bias_gat<<<(NN * GAT_OUT + TB - 1) / TB, blk, 0, stream>>>(out, b2);

  // ---- temporal branch ----
  conv1d_relu<<<dim3((LT + TB - 1) / TB, 32),  blk, 0, stream>>>(temporal, cw1, cb1, t1, CT);
  conv1d_relu<<<dim3((LT + TB - 1) / TB, 64),  blk, 0, stream>>>(t1,       cw2, cb2, t2, 32);
  conv1d_relu<<<dim3((LT + TB - 1) / TB, 128), blk, 0, stream>>>(t2,       cw3, cb3, t3, 64);

  // u = relu(t @ fw1^T + fb1)   (20000x1280x256, LDS-tiled fp32 WMMA, NT)
  wmma_gemm_nt_lds<<<dim3(NN / BM, 256 / BN), tblk, 0, stream>>>(
      t3, 1280, fw1, 1280, u, 256, fb1, 1, 1280);
  // out[:,512:576] = u @ fw2^T + fb2   (20000x256x64, LDS-tiled fp32 WMMA, NT)
  wmma_gemm_nt_lds<<<dim3(NN / BM, 64 / BN), tblk, 0, stream>>>(
      u, 256, fw2, 256, out + GAT_OUT, OUT_C, fb2, 0, 256);
}